// GraphSAGELayer_13039520710794
// MI455X (gfx1250) — compile-verified
//
#include <hip/hip_runtime.h>

typedef __attribute__((ext_vector_type(2))) float v2f;
typedef __attribute__((ext_vector_type(8))) float v8f;

#define N_NODES 100000
#define N_EDGES 1250000
#define FEATS   64
#define NTILES  (N_NODES / 16)   // 6250, exact

// ---------------- Kernel 0: zero the ah accumulator (float4 stores) --------
__global__ void GraphSAGE_zero(float4* __restrict__ ws, int n4) {
    int i = blockIdx.x * blockDim.x + threadIdx.x;
    if (i < n4) ws[i] = make_float4(0.f, 0.f, 0.f, 0.f);
}

// ---------------- Kernel 1: edge-parallel gather + scatter-add -------------
// 16 lanes per edge; each lane moves a float4 (16B) of the 256B feature row.
// h-row reads are fully coalesced; atomics land in L2 (ah fits in 192MB L2).
__global__ void GraphSAGE_scatter(const float* __restrict__ h,
                                  const long long* __restrict__ src,
                                  const long long* __restrict__ dst,
                                  float* __restrict__ ah) {
    long long tid = (long long)blockIdx.x * blockDim.x + threadIdx.x;
    long long e = tid >> 4;
    int c = (int)(tid & 15);
    if (e >= N_EDGES) return;
    int s = (int)src[e];
    int d = (int)dst[e];
    const float4 v = *(const float4*)(h + (long long)s * FEATS + c * 4);
    float* o = ah + (long long)d * FEATS + c * 4;
    atomicAdd(o + 0, v.x);
    atomicAdd(o + 1, v.y);
    atomicAdd(o + 2, v.z);
    atomicAdd(o + 3, v.w);
}

// ---------------- Kernel 2: out = relu(ah @ W.T + b) via fp32 WMMA ---------
// One wave handles one 16-row tile of ah. A-matrix (16x4 f32) layout:
//   lanes 0-15 : M=lane,  {VGPR0,VGPR1} = K={0,1}
//   lanes 16-31: M=lane-16,{VGPR0,VGPR1} = K={2,3}
// B-matrix (4x16) mirrored: lanes hold N, half-wave selects K pair.
// Both map to contiguous float2 loads. C/D: VGPR i -> M = i + 8*(lane>=16).
__global__ void __launch_bounds__(256)
GraphSAGE_gemm(const float* __restrict__ ah,
               const float* __restrict__ W,
               const float* __restrict__ bias,
               float* __restrict__ out) {
    const int lane = threadIdx.x & 31;
    const int wave = threadIdx.x >> 5;
    const int tile = blockIdx.x * 8 + wave;      // wave-uniform
    if (tile >= NTILES) return;                  // uniform exit: EXEC stays all-1 for WMMA

    const int m  = lane & 15;
    const int hi = lane >> 4;                    // 0: K pair {0,1}, 1: K pair {2,3}
    const long long row = (long long)tile * 16 + m;

    // Preload this lane's A slices for all 16 k-steps (K=64, 4 per WMMA).
    v2f areg[16];
    const float* arow = ah + row * FEATS + hi * 2;
#pragma unroll
    for (int k = 0; k < 16; ++k) {
        const float2 t = *(const float2*)(arow + k * 4);
        areg[k].x = t.x;
        areg[k].y = t.y;
    }

#pragma unroll
    for (int nt = 0; nt < 4; ++nt) {
        const int n = nt * 16 + m;               // output column this lane owns
        const float* wrow = W + n * FEATS + hi * 2;  // B[k][n] == W[n][k]
        v8f acc = {};
#pragma unroll
        for (int k = 0; k < 16; ++k) {
            v2f breg;
            const float2 t = *(const float2*)(wrow + k * 4);
            breg.x = t.x;
            breg.y = t.y;
            // 8 args: (neg_a, A, neg_b, B, c_mod, C, reuse_a, reuse_b)
            acc = __builtin_amdgcn_wmma_f32_16x16x4_f32(
                false, areg[k], false, breg, (short)0, acc, false, false);
        }
        const float bv = bias[n];
#pragma unroll
        for (int i = 0; i < 8; ++i) {
            const int orow = tile * 16 + i + hi * 8;
            float v = acc[i] + bv;
            v = v > 0.f ? v : 0.f;
            out[(long long)orow * FEATS + n] = v;
        }
    }
}

extern "C" void kernel_launch(void* const* d_in, const int* in_sizes, int n_in,
                              void* d_out, int out_size, void* d_ws, size_t ws_size,
                              hipStream_t stream) {
    const float*     h    = (const float*)d_in[0];
    const long long* src  = (const long long*)d_in[1];
    const long long* dst  = (const long long*)d_in[2];
    const float*     W    = (const float*)d_in[3];
    const float*     bias = (const float*)d_in[4];
    float* out = (float*)d_out;
    float* ah  = (float*)d_ws;   // N_NODES * FEATS floats = 25.6 MB scratch

    // 1) zero accumulator
    const int n4 = N_NODES * FEATS / 4;
    GraphSAGE_zero<<<(n4 + 255) / 256, 256, 0, stream>>>((float4*)ah, n4);

    // 2) edge scatter: 16 lanes per edge
    const long long nthreads = (long long)N_EDGES * 16;
    const int nblocks = (int)((nthreads + 255) / 256);
    GraphSAGE_scatter<<<nblocks, 256, 0, stream>>>(h, src, dst, ah);

    // 3) fused GEMM + bias + relu (8 waves / block, 1 tile of 16 nodes per wave)
    GraphSAGE_gemm<<<(NTILES + 7) / 8, 256, 0, stream>>>(ah, W, bias, out);
}